// BilinearInterpolation_54451595379233
// MI455X (gfx1250) — compile-verified
//
#include <hip/hip_runtime.h>
#include <cstdint>
#include <cstddef>

// ---------------------------------------------------------------------------
// Bicubic (Catmull-Rom) affine warp, B=8, H=W=1024, C=1, fp32.
// Memory/gather-bound: min traffic 64 MB -> ~2.8us @ 23.3 TB/s. WMMA is
// inapplicable (per-pixel data-dependent 4-tap weights, no shared operand).
// CDNA5 path used: async global->LDS staging (ASYNCcnt) of a 48x48
// edge-replicated halo tile + s_wait_asynccnt.
// H,W are compile-time constants so the linspace steps are literals (no fp32
// division in the prologue) and row strides strength-reduce to shifts.
// ---------------------------------------------------------------------------

#define TILE 32
#define HALO 8
#define LDSW (TILE + 2 * HALO)   // 48
#define NLDS (LDSW * LDSW)       // 2304 floats = 9216 B LDS

constexpr int kH = 1024;
constexpr int kW = 1024;

// ---- CDNA5 async global->LDS copy (gfx1250) --------------------------------
#if defined(__has_builtin)
#  if __has_builtin(__builtin_amdgcn_global_load_async_to_lds_b128)
#    define HAVE_ASYNC_B128 1
#  endif
#  if __has_builtin(__builtin_amdgcn_s_wait_asynccnt)
#    define HAVE_WAIT_ASYNC 1
#  endif
#endif

typedef int v4i_t __attribute__((vector_size(16)));
typedef __attribute__((address_space(1))) v4i_t* g4p_t;  // global int4*
typedef __attribute__((address_space(3))) v4i_t* l4p_t;  // LDS    int4*

__device__ __forceinline__ void async_copy_b128(const float* gsrc, float* ldst) {
#if defined(HAVE_ASYNC_B128)
  __builtin_amdgcn_global_load_async_to_lds_b128(
      (g4p_t)const_cast<float*>(gsrc),
      (l4p_t)ldst,
      /*offset=*/0, /*cpol=*/0);
#else
  // LDS aperture: low 32 bits of a generic LDS pointer are the LDS byte offset.
  const unsigned lds_off = (unsigned)(uintptr_t)ldst;
  const unsigned long long ga = (unsigned long long)(uintptr_t)gsrc;
  asm volatile("global_load_async_to_lds_b128 %0, %1, off"
               :: "v"(lds_off), "v"(ga)
               : "memory");
#endif
}

__device__ __forceinline__ void wait_async_zero() {
#if defined(HAVE_WAIT_ASYNC)
  __builtin_amdgcn_s_wait_asynccnt(0);
#else
  asm volatile("s_wait_asynccnt 0" ::: "memory");
#endif
}

// Catmull-Rom weights; algebraically identical to the reference Hermite form.
__device__ __forceinline__ void cr_weights(float t, float& w0, float& w1,
                                           float& w2, float& w3) {
  w0 = t * fmaf(t, fmaf(-0.5f, t, 1.0f), -0.5f);          // -0.5t^3 +  t^2 - 0.5t
  w1 = fmaf(t * t, fmaf(1.5f, t, -2.5f), 1.0f);           //  1.5t^3 - 2.5t^2 + 1
  w2 = t * fmaf(t, fmaf(-1.5f, t, 2.0f), 0.5f);           // -1.5t^3 + 2t^2 + 0.5t
  w3 = t * t * fmaf(0.5f, t, -0.5f);                      //  0.5t^3 - 0.5t^2
}

__device__ __forceinline__ int iclamp(int v, int lo, int hi) {
  return v < lo ? lo : (v > hi ? hi : v);
}

__global__ __launch_bounds__(256) void bicubic_warp_kernel(
    const float* __restrict__ X, const float* __restrict__ theta,
    float* __restrict__ out) {
  constexpr int H = kH, W = kW;
  constexpr float sx = 2.0f / (float)(W - 1);   // linspace(-1,1,W) step
  constexpr float sy = 2.0f / (float)(H - 1);

  __shared__ float tile[NLDS];

  const int b  = blockIdx.z;
  const int x0 = blockIdx.x * TILE;
  const int y0 = blockIdx.y * TILE;
  const int tx = (int)threadIdx.x;   // 0..31 (one wave32 per row)
  const int ty = (int)threadIdx.y;   // 0..7
  const int tid = ty * 32 + tx;

  const float* th = theta + (size_t)b * 6;
  const float ax = th[0] - 1.0f, bx = th[1], cx = th[2];  // fx = ax*xn+bx*yn+cx
  const float ay = th[3], by = th[4] - 1.0f, cy = th[5];  // fy = ay*xn+by*yn+cy

  // Per-tile displacement bounds: affine => extremes at the 4 tile corners.
  const float xnL = fmaf((float)x0, sx, -1.0f);
  const float xnR = fmaf((float)(x0 + TILE - 1), sx, -1.0f);
  const float ynT = fmaf((float)y0, sy, -1.0f);
  const float ynB = fmaf((float)(y0 + TILE - 1), sy, -1.0f);

  const float fxA = fmaf(ax, xnL, fmaf(bx, ynT, cx));
  const float fxB = fmaf(ax, xnR, fmaf(bx, ynT, cx));
  const float fxC = fmaf(ax, xnL, fmaf(bx, ynB, cx));
  const float fxD = fmaf(ax, xnR, fmaf(bx, ynB, cx));
  const float fyA = fmaf(ay, xnL, fmaf(by, ynT, cy));
  const float fyB = fmaf(ay, xnR, fmaf(by, ynT, cy));
  const float fyC = fmaf(ay, xnL, fmaf(by, ynB, cy));
  const float fyD = fmaf(ay, xnR, fmaf(by, ynB, cy));

  const int dxmin = (int)floorf(fminf(fminf(fxA, fxB), fminf(fxC, fxD)));
  const int dxmax = (int)floorf(fmaxf(fmaxf(fxA, fxB), fmaxf(fxC, fxD)));
  const int dymin = (int)floorf(fminf(fminf(fyA, fyB), fminf(fyC, fyD)));
  const int dymax = (int)floorf(fmaxf(fmaxf(fyA, fyB), fmaxf(fyC, fyD)));

  // Halo covers dx in [-(HALO-1), HALO-2]; keep one extra cell of margin for
  // fma rounding differences between corner and per-pixel evaluation.
  const bool fits = (dxmin >= -(HALO - 2)) && (dxmax <= (HALO - 3)) &&
                    (dymin >= -(HALO - 2)) && (dymax <= (HALO - 3));

  const int x = x0 + tx;
  const float xn = fmaf((float)x, sx, -1.0f);
  constexpr size_t img = (size_t)H * (size_t)W;
  const float* __restrict__ Xb = X + (size_t)b * img;

  if (fits) {
    // ---- stage 48x48 halo (edge-replicated) into LDS ----
    const bool interior = (x0 >= HALO) && (x0 + TILE + HALO <= W) &&
                          (y0 >= HALO) && (y0 + TILE + HALO <= H);
    if (interior) {
      // 48 rows x 12 float4 = 576 async b128 copies; 16B aligned on both ends.
      // 576 / 256 threads = 2 copies each + 1 extra for tid < 64 (peeled).
      const float* gorg = Xb + (size_t)(y0 - HALO) * W + (x0 - HALO);
      {
        const int r0 = tid / (LDSW / 4), c0 = (tid % (LDSW / 4)) * 4;
        async_copy_b128(gorg + (size_t)r0 * W + c0, &tile[r0 * LDSW + c0]);
        const int i1 = tid + 256;
        const int r1 = i1 / (LDSW / 4), c1 = (i1 % (LDSW / 4)) * 4;
        async_copy_b128(gorg + (size_t)r1 * W + c1, &tile[r1 * LDSW + c1]);
        const int i2 = tid + 512;
        if (i2 < LDSW * (LDSW / 4)) {
          const int r2 = i2 / (LDSW / 4), c2 = (i2 % (LDSW / 4)) * 4;
          async_copy_b128(gorg + (size_t)r2 * W + c2, &tile[r2 * LDSW + c2]);
        }
      }
      wait_async_zero();
      __syncthreads();
    } else {
      // Border tile: clamped scalar staging (replicates edges, matching clip()).
      for (int i = tid; i < NLDS; i += 256) {
        const int r = i / LDSW, c = i % LDSW;
        const int gr = iclamp(y0 - HALO + r, 0, H - 1);
        const int gc = iclamp(x0 - HALO + c, 0, W - 1);
        tile[i] = Xb[(size_t)gr * W + gc];
      }
      __syncthreads();
    }

    // ---- compute 4 output rows per thread from LDS ----
#pragma unroll
    for (int k = 0; k < 4; ++k) {
      const int ly = ty + k * 8;
      const int y  = y0 + ly;
      const float yn = fmaf((float)y, sy, -1.0f);
      const float fx = fmaf(ax, xn, fmaf(bx, yn, cx));
      const float fy = fmaf(ay, xn, fmaf(by, yn, cy));
      const float ffx = floorf(fx), ffy = floorf(fy);
      const float xt = fx - ffx, yt = fy - ffy;
      const int dx = (int)ffx, dy = (int)ffy;

      float wx0, wx1, wx2, wx3, wy0, wy1, wy2, wy3;
      cr_weights(xt, wx0, wx1, wx2, wx3);
      cr_weights(yt, wy0, wy1, wy2, wy3);

      const int hx = tx + HALO + dx - 1;     // in [1,46] by `fits`
      const int hy = ly + HALO + dy - 1;
      const float* base = &tile[hy * LDSW + hx];

      float col[4];
#pragma unroll
      for (int i = 0; i < 4; ++i) {
        const float v0 = base[i * LDSW + 0];
        const float v1 = base[i * LDSW + 1];
        const float v2 = base[i * LDSW + 2];
        const float v3 = base[i * LDSW + 3];
        col[i] = fmaf(wx0, v0, fmaf(wx1, v1, fmaf(wx2, v2, wx3 * v3)));
      }
      const float r = fmaf(wy0, col[0],
                      fmaf(wy1, col[1], fmaf(wy2, col[2], wy3 * col[3])));
      out[(size_t)b * img + (size_t)y * W + x] = r;
    }
  } else {
    // ---- general fallback (displacement exceeds halo): clamped global gather
#pragma unroll
    for (int k = 0; k < 4; ++k) {
      const int ly = ty + k * 8;
      const int y  = y0 + ly;
      const float yn = fmaf((float)y, sy, -1.0f);
      const float fx = fmaf(ax, xn, fmaf(bx, yn, cx));
      const float fy = fmaf(ay, xn, fmaf(by, yn, cy));
      const float ffx = floorf(fx), ffy = floorf(fy);
      const float xt = fx - ffx, yt = fy - ffy;
      const int dx = (int)ffx, dy = (int)ffy;

      float wx0, wx1, wx2, wx3, wy0, wy1, wy2, wy3;
      cr_weights(xt, wx0, wx1, wx2, wx3);
      cr_weights(yt, wy0, wy1, wy2, wy3);

      const int xb = x + dx - 1, yb = y + dy - 1;
      const int xi0 = iclamp(xb + 0, 0, W - 1);
      const int xi1 = iclamp(xb + 1, 0, W - 1);
      const int xi2 = iclamp(xb + 2, 0, W - 1);
      const int xi3 = iclamp(xb + 3, 0, W - 1);

      float col[4];
#pragma unroll
      for (int i = 0; i < 4; ++i) {
        const int yi = iclamp(yb + i, 0, H - 1);
        const float* row = Xb + (size_t)yi * W;
        col[i] = fmaf(wx0, row[xi0],
                 fmaf(wx1, row[xi1],
                 fmaf(wx2, row[xi2], wx3 * row[xi3])));
      }
      const float r = fmaf(wy0, col[0],
                      fmaf(wy1, col[1], fmaf(wy2, col[2], wy3 * col[3])));
      out[(size_t)b * img + (size_t)y * W + x] = r;
    }
  }
}

extern "C" void kernel_launch(void* const* d_in, const int* in_sizes, int n_in,
                              void* d_out, int out_size, void* d_ws,
                              size_t ws_size, hipStream_t stream) {
  (void)n_in; (void)out_size; (void)d_ws; (void)ws_size;
  const float* X     = (const float*)d_in[0];
  const float* theta = (const float*)d_in[1];
  float* out = (float*)d_out;

  const int B = in_sizes[0] / (kH * kW);        // 8

  dim3 block(32, 8, 1);                          // 8 x wave32
  dim3 grid(kW / TILE, kH / TILE, B);
  bicubic_warp_kernel<<<grid, block, 0, stream>>>(X, theta, out);
}